// GINLayer_53755810677327
// MI455X (gfx1250) — compile-verified
//
#include <hip/hip_runtime.h>

#define N_NODES 100000
#define N_EDGES 1600000
#define CH 128
#define BN_EPS 1e-5f

typedef __attribute__((ext_vector_type(2))) float v2f;
typedef __attribute__((ext_vector_type(8))) float v8f;

// ---------------------------------------------------------------------------
// Kernel 1: agg = x (GIN eps=0 self term), zero BN accumulators
// ---------------------------------------------------------------------------
__global__ void k_init(const float* __restrict__ x, float* __restrict__ agg,
                       float* __restrict__ stats) {
  size_t i = (size_t)blockIdx.x * blockDim.x + threadIdx.x;
  size_t n4 = (size_t)N_NODES * CH / 4;
  if (i < n4) ((float4*)agg)[i] = ((const float4*)x)[i];
  if (i < 2 * CH) stats[i] = 0.0f;  // sum[128], sumsq[128]
}

// ---------------------------------------------------------------------------
// Kernel 2: one wave per edge; 32 lanes x float4 = 128 channels.
// Coalesced 512B gather of x[src], atomic scatter-add into agg[dst].
// ---------------------------------------------------------------------------
__global__ __launch_bounds__(256) void k_scatter(const float* __restrict__ x,
                                                 const int* __restrict__ ei,
                                                 float* __restrict__ agg) {
  int e = blockIdx.x * (blockDim.x >> 5) + (threadIdx.x >> 5);  // wave-uniform
  if (e >= N_EDGES) return;
  int lane = threadIdx.x & 31;
  int s = ei[e];
  int d = ei[N_EDGES + e];
  float4 v = ((const float4*)(x + (size_t)s * CH))[lane];
  float* a = agg + (size_t)d * CH + lane * 4;
  atomicAdd(a + 0, v.x);
  atomicAdd(a + 1, v.y);
  atomicAdd(a + 2, v.z);
  atomicAdd(a + 3, v.w);
}

// ---------------------------------------------------------------------------
// GEMM: out[16 rows x 128 cols per wave] = op(A) @ W + bias, f32 WMMA 16x16x4.
// 8 waves per block; W staged in LDS (64 KB). BN_RELU_IN applies the fused
// BatchNorm affine + ReLU to A elements on load; RELU_OUT applied at store.
// ---------------------------------------------------------------------------
template <bool BN_RELU_IN, bool RELU_OUT>
__global__ __launch_bounds__(256) void k_gemm(const float* __restrict__ A,
                                              const float* __restrict__ W,
                                              const float* __restrict__ bias,
                                              const float* __restrict__ scale,
                                              const float* __restrict__ shift,
                                              float* __restrict__ out) {
  __shared__ float sW[CH * CH];  // 64 KB of 320 KB/WGP
  for (int i = threadIdx.x; i < CH * CH; i += 256) sW[i] = W[i];
  __syncthreads();

  const int lane = threadIdx.x & 31;
  const int wave = threadIdx.x >> 5;
  const int tile = blockIdx.x * 8 + wave;          // 16-row tile index
  if (tile * 16 >= N_NODES) return;                // wave-uniform exit
  const int row0 = tile * 16;
  const int m = lane & 15;                         // A row within tile
  const int khalf = (lane >> 4) << 1;              // lanes 16-31 hold K+2,K+3
  const int col = lane & 15;                       // B/C col within 16-tile

  v8f acc[8];
#pragma unroll
  for (int n = 0; n < 8; ++n) {
    float bv = bias[n * 16 + col];                 // acc col fixed per lane
#pragma unroll
    for (int r = 0; r < 8; ++r) acc[n][r] = bv;
  }

  const float* arow = A + (size_t)(row0 + m) * CH;
  for (int k = 0; k < CH; k += 4) {
    const int kk = k + khalf;
    v2f a;
    if (BN_RELU_IN) {
      a.x = fmaxf(arow[kk] * scale[kk] + shift[kk], 0.0f);
      a.y = fmaxf(arow[kk + 1] * scale[kk + 1] + shift[kk + 1], 0.0f);
    } else {
      a.x = arow[kk];
      a.y = arow[kk + 1];
    }
#pragma unroll
    for (int n = 0; n < 8; ++n) {
      v2f b;  // B 4x16: VGPR0 rows {k,k+2}, VGPR1 rows {k+1,k+3} by lane half
      b.x = sW[(size_t)kk * CH + n * 16 + col];
      b.y = sW[(size_t)(kk + 1) * CH + n * 16 + col];
      acc[n] = __builtin_amdgcn_wmma_f32_16x16x4_f32(
          false, a, false, b, (short)0, acc[n], false, false);
    }
  }

  // C/D layout: VGPR r -> row r (lanes 0-15) or r+8 (lanes 16-31), col fixed
  const int rbase = row0 + ((lane >> 4) << 3);
#pragma unroll
  for (int n = 0; n < 8; ++n) {
#pragma unroll
    for (int r = 0; r < 8; ++r) {
      float v = acc[n][r];
      if (RELU_OUT) v = fmaxf(v, 0.0f);
      out[(size_t)(rbase + r) * CH + n * 16 + col] = v;
    }
  }
}

// ---------------------------------------------------------------------------
// BN statistics: per-wave row ownership, register accumulation, LDS reduce,
// 256 global atomics per block into stats[0..127]=sum, stats[128..255]=sumsq.
// ---------------------------------------------------------------------------
__global__ __launch_bounds__(256) void k_bnstats(const float* __restrict__ h,
                                                 float* __restrict__ stats) {
  __shared__ float s_sum[CH];
  __shared__ float s_sq[CH];
  if (threadIdx.x < CH) {
    s_sum[threadIdx.x] = 0.0f;
    s_sq[threadIdx.x] = 0.0f;
  }
  __syncthreads();

  const int lane = threadIdx.x & 31;
  const int grp = threadIdx.x >> 5;
  float4 sum = {0.f, 0.f, 0.f, 0.f};
  float4 sq = {0.f, 0.f, 0.f, 0.f};
  for (int r = blockIdx.x * 8 + grp; r < N_NODES; r += gridDim.x * 8) {
    float4 v = ((const float4*)(h + (size_t)r * CH))[lane];
    sum.x += v.x; sum.y += v.y; sum.z += v.z; sum.w += v.w;
    sq.x += v.x * v.x; sq.y += v.y * v.y; sq.z += v.z * v.z; sq.w += v.w * v.w;
  }
  const int c = lane * 4;
  atomicAdd(&s_sum[c + 0], sum.x);
  atomicAdd(&s_sum[c + 1], sum.y);
  atomicAdd(&s_sum[c + 2], sum.z);
  atomicAdd(&s_sum[c + 3], sum.w);
  atomicAdd(&s_sq[c + 0], sq.x);
  atomicAdd(&s_sq[c + 1], sq.y);
  atomicAdd(&s_sq[c + 2], sq.z);
  atomicAdd(&s_sq[c + 3], sq.w);
  __syncthreads();
  if (threadIdx.x < CH) {
    atomicAdd(&stats[threadIdx.x], s_sum[threadIdx.x]);
    atomicAdd(&stats[CH + threadIdx.x], s_sq[threadIdx.x]);
  }
}

// ---------------------------------------------------------------------------
// Fold mean/var/gamma/beta into scale/shift: bn(x) = x*scale + shift
// ---------------------------------------------------------------------------
__global__ void k_bnfinal(const float* __restrict__ gamma,
                          const float* __restrict__ beta,
                          float* __restrict__ stats) {
  int c = threadIdx.x;
  if (c < CH) {
    float mean = stats[c] * (1.0f / N_NODES);
    float var = stats[CH + c] * (1.0f / N_NODES) - mean * mean;
    float sc = gamma[c] * rsqrtf(var + BN_EPS);
    stats[2 * CH + c] = sc;
    stats[3 * CH + c] = beta[c] - mean * sc;
  }
}

// ---------------------------------------------------------------------------
extern "C" void kernel_launch(void* const* d_in, const int* in_sizes, int n_in,
                              void* d_out, int out_size, void* d_ws,
                              size_t ws_size, hipStream_t stream) {
  const float* x = (const float*)d_in[0];
  const int* ei = (const int*)d_in[1];
  const float* W1 = (const float*)d_in[2];
  const float* b1 = (const float*)d_in[3];
  const float* gamma = (const float*)d_in[4];
  const float* beta = (const float*)d_in[5];
  const float* W2 = (const float*)d_in[6];
  const float* b2 = (const float*)d_in[7];
  float* out = (float*)d_out;

  char* ws = (char*)d_ws;
  float* stats = (float*)ws;  // [4*128]: sum, sumsq, scale, shift
  float* agg = (float*)(ws + 4096);
  float* h1 = (float*)(ws + 4096 + (size_t)N_NODES * CH * sizeof(float));

  const int n4 = N_NODES * CH / 4;  // 3.2M float4
  k_init<<<(n4 + 255) / 256, 256, 0, stream>>>(x, agg, stats);

  k_scatter<<<(N_EDGES + 7) / 8, 256, 0, stream>>>(x, ei, agg);

  const int tiles = N_NODES / 16;               // 6250 (exact)
  const int gblocks = (tiles + 7) / 8;          // 8 waves/block
  k_gemm<false, false><<<gblocks, 256, 0, stream>>>(agg, W1, b1, nullptr,
                                                    nullptr, h1);

  k_bnstats<<<240, 256, 0, stream>>>(h1, stats);
  k_bnfinal<<<1, 128, 0, stream>>>(gamma, beta, stats);

  k_gemm<true, true><<<gblocks, 256, 0, stream>>>(h1, W2, b2, stats + 2 * CH,
                                                  stats + 3 * CH, out);
}